// MoE_4707284156658
// MI455X (gfx1250) — compile-verified
//
#include <hip/hip_runtime.h>

// ---------------------------------------------------------------------------
// MoE (top-2 of 8 experts) for MI455X / gfx1250, wave32 + WMMA f32 16x16x4.
// Only the 2 selected experts per batch are computed (gates of others == 0).
// ---------------------------------------------------------------------------

typedef float v2f __attribute__((ext_vector_type(2)));
typedef float v4f __attribute__((ext_vector_type(4)));
typedef float v8f __attribute__((ext_vector_type(8)));

#define B_ 8
#define S_ 2048
#define D_ 1024
#define O_ 1024
#define E_ 8

#define BS 128          // s-rows per block
#define BO 64           // o-cols per block
#define KT 16           // K depth per pipeline stage
#define NKT (D_ / KT)   // 64 stages
#define LSTR 20         // LDS row stride (floats): 80B rows -> 16B aligned, conflict-free

// ---- optional CDNA5 async global->LDS path (ASYNCcnt) ---------------------
#if defined(__has_builtin)
#  if __has_builtin(__builtin_amdgcn_global_load_async_to_lds_b128)
#    define HAVE_ASYNC_LOAD 1
#  endif
#  if __has_builtin(__builtin_amdgcn_s_wait_asynccnt)
#    define HAVE_WAIT_ASYNC 1
#  endif
#endif
#ifndef HAVE_ASYNC_LOAD
#  define HAVE_ASYNC_LOAD 0
#endif
#ifndef HAVE_WAIT_ASYNC
#  define HAVE_WAIT_ASYNC 0
#endif

// Exact parameter types per the compiler diagnostic:
//   param0: 'int __attribute__((vector_size(16))) __device__ *'  (global / AS1)
//   param1: LDS side (AS3), same pointee
typedef int v4i_vs __attribute__((vector_size(4 * sizeof(int))));
typedef __attribute__((address_space(1))) v4i_vs gv4i;
typedef __attribute__((address_space(3))) v4i_vs lv4i;

__device__ __forceinline__ void wait_async_zero() {
#if HAVE_WAIT_ASYNC
    __builtin_amdgcn_s_wait_asynccnt(0);
#else
    asm volatile("s_wait_asynccnt 0x0" ::: "memory");
#endif
}

__device__ __forceinline__ float gelu_exact(float v) {
    return 0.5f * v * (1.0f + erff(v * 0.70710678118654752440f));
}

// ---------------------------------------------------------------------------
// K1: pooled[b][d] = mean_s x[b][s][d]
// ---------------------------------------------------------------------------
__global__ __launch_bounds__(256) void pooled_kernel(const float* __restrict__ x,
                                                     float* __restrict__ pooled) {
    const int b = blockIdx.x;
    const int d = blockIdx.y * 256 + threadIdx.x;
    const float* p = x + (size_t)b * S_ * D_ + d;
    float acc = 0.0f;
#pragma unroll 8
    for (int s = 0; s < S_; ++s) acc += p[(size_t)s * D_];
    pooled[b * D_ + d] = acc * (1.0f / (float)S_);
}

// ---------------------------------------------------------------------------
// K2: logits[b][e] = pooled[b] . gate_w[e] + gate_b[e]
// ---------------------------------------------------------------------------
__global__ __launch_bounds__(256) void logits_kernel(const float* __restrict__ pooled,
                                                     const float* __restrict__ gate_w,
                                                     const float* __restrict__ gate_b,
                                                     float* __restrict__ logits) {
    __shared__ float red[256];
    const int b = blockIdx.x, e = blockIdx.y;
    const float* pp = pooled + b * D_;
    const float* gw = gate_w + e * D_;
    float acc = 0.0f;
    for (int d = threadIdx.x; d < D_; d += 256) acc += pp[d] * gw[d];
    red[threadIdx.x] = acc;
    __syncthreads();
    for (int st = 128; st > 0; st >>= 1) {
        if (threadIdx.x < st) red[threadIdx.x] += red[threadIdx.x + st];
        __syncthreads();
    }
    if (threadIdx.x == 0) logits[b * E_ + e] = red[0] + gate_b[e];
}

// ---------------------------------------------------------------------------
// K3: per-batch top-2 + softmax over the two survivors
// ---------------------------------------------------------------------------
__global__ void topk_kernel(const float* __restrict__ logits,
                            float* __restrict__ gates, int* __restrict__ eidx) {
    const int b = threadIdx.x;
    if (b >= B_) return;
    float l[E_];
#pragma unroll
    for (int e = 0; e < E_; ++e) l[e] = logits[b * E_ + e];
    int i1 = 0;
#pragma unroll
    for (int e = 1; e < E_; ++e) if (l[e] > l[i1]) i1 = e;
    int i2 = (i1 == 0) ? 1 : 0;
#pragma unroll
    for (int e = 0; e < E_; ++e) if (e != i1 && l[e] > l[i2]) i2 = e;
    const float m  = l[i1];
    const float x1 = __expf(l[i1] - m);
    const float x2 = __expf(l[i2] - m);
    const float inv = 1.0f / (x1 + x2);
    gates[b * 2 + 0] = x1 * inv;
    gates[b * 2 + 1] = x2 * inv;
    eidx[b * 2 + 0] = i1;
    eidx[b * 2 + 1] = i2;
}

// ---------------------------------------------------------------------------
// K4: out[b] = g0*gelu(X W_e0^T + b_e0) + g1*gelu(X W_e1^T + b_e1)
//  block: 256 threads = 8 waves; block tile 128s x 64o; wave tile 32s x 32o
//  per wave: 2(experts) x 2(s) x 2(o) f32 16x16 accumulators, K step 16.
// ---------------------------------------------------------------------------
__global__ __launch_bounds__(256) void moe_gemm_kernel(const float* __restrict__ x,
                                                       const float* __restrict__ expert_w,
                                                       const float* __restrict__ expert_b,
                                                       const float* __restrict__ gates,
                                                       const int* __restrict__ eidx,
                                                       float* __restrict__ out) {
    __shared__ float lds_x[2][BS * LSTR];       // 2 * 128*20*4B = 20 KB
    __shared__ float lds_w[2][2 * BO * LSTR];   // 2 * 128*20*4B = 20 KB

    const int b    = blockIdx.y;
    const int nOT  = O_ / BO;                    // 16
    const int o0   = (blockIdx.x % nOT) * BO;
    const int s0   = (blockIdx.x / nOT) * BS;

    const int tid  = threadIdx.x;
    const int lane = tid & 31;
    const int wid  = tid >> 5;                   // 0..7
    const int lm   = lane & 15;                  // row/col within fragment
    const int kh   = lane >> 4;                  // K-half select (f32 A/B layout)
    const int wS   = wid >> 1;                   // 0..3  (32-row band)
    const int wO   = wid & 1;                    // 0..1  (32-col band)

    const int   e0 = eidx[b * 2 + 0];
    const int   e1 = eidx[b * 2 + 1];
    const float g0 = gates[b * 2 + 0];
    const float g1 = gates[b * 2 + 1];

    const float* xb = x + (size_t)b * S_ * D_;
    const float* w0 = expert_w + (size_t)e0 * O_ * D_;
    const float* w1 = expert_w + (size_t)e1 * O_ * D_;

    // staging coordinates: each thread moves 2 float4 of X and 2 float4 of W
    int xrow[2], xcg[2], wrow[2], wcg[2], whalf[2];
#pragma unroll
    for (int i = 0; i < 2; ++i) {
        const int xi = tid + i * 256;            // 0..511 -> 128 rows x 4 groups
        xrow[i] = xi >> 2; xcg[i] = xi & 3;
        const int wi  = tid + i * 256;           // 0..511 -> 2 experts x 64 rows x 4 groups
        whalf[i] = wi >> 8;
        const int rem = wi & 255;
        wrow[i] = rem >> 2; wcg[i] = rem & 3;
    }

    v4f xreg[2], wreg[2];

    auto g_load = [&](int kt) {
        const int k0 = kt * KT;
#pragma unroll
        for (int i = 0; i < 2; ++i) {
            xreg[i] = *(const v4f*)(xb + (size_t)(s0 + xrow[i]) * D_ + k0 + xcg[i] * 4);
            const float* wp = whalf[i] ? w1 : w0;
            wreg[i] = *(const v4f*)(wp + (size_t)(o0 + wrow[i]) * D_ + k0 + wcg[i] * 4);
        }
    };
    auto l_store = [&](int buf) {
#pragma unroll
        for (int i = 0; i < 2; ++i) {
            *(v4f*)&lds_x[buf][xrow[i] * LSTR + xcg[i] * 4] = xreg[i];
            *(v4f*)&lds_w[buf][(whalf[i] * BO + wrow[i]) * LSTR + wcg[i] * 4] = wreg[i];
        }
    };
#if HAVE_ASYNC_LOAD
    auto a_load = [&](int kt, int buf) {
        const int k0 = kt * KT;
#pragma unroll
        for (int i = 0; i < 2; ++i) {
            const float* gx = xb + (size_t)(s0 + xrow[i]) * D_ + k0 + xcg[i] * 4;
            float* lx = &lds_x[buf][xrow[i] * LSTR + xcg[i] * 4];
            __builtin_amdgcn_global_load_async_to_lds_b128((gv4i*)gx, (lv4i*)lx, 0, 0);
            const float* wp = whalf[i] ? w1 : w0;
            const float* gw = wp + (size_t)(o0 + wrow[i]) * D_ + k0 + wcg[i] * 4;
            float* lw = &lds_w[buf][(whalf[i] * BO + wrow[i]) * LSTR + wcg[i] * 4];
            __builtin_amdgcn_global_load_async_to_lds_b128((gv4i*)gw, (lv4i*)lw, 0, 0);
        }
    };
#endif

    v8f acc[2][2][2];   // [expert][sTile][oTile]
#pragma unroll
    for (int e = 0; e < 2; ++e)
#pragma unroll
        for (int i = 0; i < 2; ++i)
#pragma unroll
            for (int j = 0; j < 2; ++j) acc[e][i][j] = 0.0f;

    auto compute = [&](int buf) {
        const float* lx = lds_x[buf];
        const float* lw = lds_w[buf];
#pragma unroll
        for (int kk = 0; kk < 4; ++kk) {
            const int kc = kk * 4 + kh * 2;      // K pair per f32 WMMA A/B layout
            v2f a[2];
#pragma unroll
            for (int i = 0; i < 2; ++i)
                a[i] = *(const v2f*)&lx[(wS * 32 + i * 16 + lm) * LSTR + kc];
#pragma unroll
            for (int e = 0; e < 2; ++e)
#pragma unroll
                for (int j = 0; j < 2; ++j) {
                    v2f bf = *(const v2f*)&lw[(e * BO + wO * 32 + j * 16 + lm) * LSTR + kc];
#pragma unroll
                    for (int i = 0; i < 2; ++i)
                        acc[e][i][j] = __builtin_amdgcn_wmma_f32_16x16x4_f32(
                            false, a[i], false, bf, (short)0, acc[e][i][j], false, false);
                }
        }
    };

    // ------------------ double-buffered K pipeline ------------------
#if HAVE_ASYNC_LOAD
    a_load(0, 0);
    wait_async_zero();
    __syncthreads();
    for (int kt = 0; kt < NKT; ++kt) {
        const int cur = kt & 1;
        if (kt + 1 < NKT) a_load(kt + 1, cur ^ 1);
        compute(cur);
        if (kt + 1 < NKT) wait_async_zero();
        __syncthreads();
    }
#else
    g_load(0);
    l_store(0);
    __syncthreads();
    for (int kt = 0; kt < NKT; ++kt) {
        const int cur = kt & 1;
        if (kt + 1 < NKT) g_load(kt + 1);
        compute(cur);
        if (kt + 1 < NKT) l_store(cur ^ 1);
        __syncthreads();
    }
#endif

    // ------------------ epilogue: bias + exact GELU + gate mix ------------------
    float* ob = out + (size_t)b * S_ * O_;
#pragma unroll
    for (int i = 0; i < 2; ++i)
#pragma unroll
        for (int j = 0; j < 2; ++j) {
            const int col   = o0 + wO * 32 + j * 16 + lm;   // N fixed per lane across the 8 regs
            const float bb0 = expert_b[e0 * O_ + col];
            const float bb1 = expert_b[e1 * O_ + col];
            const int rbase = s0 + wS * 32 + i * 16 + kh * 8;
#pragma unroll
            for (int v = 0; v < 8; ++v) {
                const float z0 = acc[0][i][j][v] + bb0;
                const float z1 = acc[1][i][j][v] + bb1;
                ob[(size_t)(rbase + v) * O_ + col] = g0 * gelu_exact(z0) + g1 * gelu_exact(z1);
            }
        }
}

// ---------------------------------------------------------------------------
extern "C" void kernel_launch(void* const* d_in, const int* in_sizes, int n_in,
                              void* d_out, int out_size, void* d_ws, size_t ws_size,
                              hipStream_t stream) {
    const float* x        = (const float*)d_in[0];
    const float* gate_w   = (const float*)d_in[1];
    const float* gate_b   = (const float*)d_in[2];
    const float* expert_w = (const float*)d_in[3];
    const float* expert_b = (const float*)d_in[4];
    float*       out      = (float*)d_out;

    float* ws_pooled = (float*)d_ws;                 // B*D floats
    float* ws_logits = ws_pooled + B_ * D_;          // B*E floats
    float* ws_gates  = ws_logits + B_ * E_;          // B*2 floats
    int*   ws_eidx   = (int*)(ws_gates + B_ * 2);    // B*2 ints

    pooled_kernel<<<dim3(B_, D_ / 256), 256, 0, stream>>>(x, ws_pooled);
    logits_kernel<<<dim3(B_, E_), 256, 0, stream>>>(ws_pooled, gate_w, gate_b, ws_logits);
    topk_kernel<<<1, 32, 0, stream>>>(ws_logits, ws_gates, ws_eidx);

    const int nTiles = (S_ / BS) * (O_ / BO);        // 16 * 16 = 256
    moe_gemm_kernel<<<dim3(nTiles, B_), 256, 0, stream>>>(x, expert_w, expert_b,
                                                          ws_gates, ws_eidx, out);
}